// RoPeBruteForce_2637109920100
// MI455X (gfx1250) — compile-verified
//
#include <hip/hip_runtime.h>
#include <cstdint>

// out[b,h,s,d] = diag(R)[d,s] * x[b,h,s,d]
// Pure streaming elementwise op: ~514 MB HBM traffic -> ~22us @ 23.3 TB/s.
// CDNA5 path: TDM tensor_load_to_lds stages the 2MB diagonal slice (128x64
// tile per block, LDS-padded), NT-hinted b128 streams for x/out.

typedef float    f32x4 __attribute__((ext_vector_type(4)));
typedef uint32_t u32x4 __attribute__((ext_vector_type(4)));
typedef int      i32x4 __attribute__((ext_vector_type(4)));
typedef int      i32x8 __attribute__((ext_vector_type(8)));

#define SEQ       4096
#define DK        128
#define S_CHUNK   64
#define BH_CHUNK  8
#define ROW_PITCH 65   // 64 data DWORDs + 1 TDM pad DWORD per row (bank-conflict pad)

__global__ __launch_bounds__(256) void rope_diag_scale_kernel(
    const float* __restrict__ x, const float* __restrict__ R,
    float* __restrict__ out)
{
    __shared__ float coeff[DK * ROW_PITCH];   // 33,280 B

    const int tid = threadIdx.x;
    const int s0  = blockIdx.x * S_CHUNK;     // seq chunk
    const int bh0 = blockIdx.y * BH_CHUNK;    // (b*16+h) chunk

    // ---- TDM: DMA the 128(d) x 64(s) diagonal-coefficient tile into LDS ----
    // diag(R)[d][s] = R[d*129*4096 + s]: rows contiguous in s, row stride
    // 129*4096 elements. One wave issues the tensor DMA; EXEC is ignored.
    if (tid < 32) {
        const uint64_t gaddr = (uint64_t)(uintptr_t)R + (uint64_t)s0 * 4u;
        const uint32_t lds   = (uint32_t)(uintptr_t)&coeff[0]; // low 32b = LDS byte addr

        u32x4 g0;
        g0[0] = 1u;                                        // count=1, user descriptor
        g0[1] = lds;                                       // lds_addr
        g0[2] = (uint32_t)(gaddr & 0xFFFFFFFFu);           // global_addr[31:0]
        g0[3] = (uint32_t)((gaddr >> 32) & 0x1FFFFFFu)     // global_addr[56:32]
              | (2u << 30);                                // type = 2 ("image")

        i32x8 g1;
        g1[0] = (2 << 16)                                  // data_size = 4B
              | (1 << 20)                                  // pad_enable (LDS dest)
              | (5 << 22);                                 // pad_interval=64dw, amount=1dw
        g1[1] = (SEQ & 0xFFFF) << 16;                      // tensor_dim0 lo16 (=4096)
        g1[2] = ((SEQ >> 16) & 0xFFFF)                     // tensor_dim0 hi16
              | ((DK & 0xFFFF) << 16);                     // tensor_dim1 lo16 (=128)
        g1[3] = ((DK >> 16) & 0xFFFF)                      // tensor_dim1 hi16
              | (S_CHUNK << 16);                           // tile_dim0 = 64
        g1[4] = DK;                                        // tile_dim1=128, tile_dim2=0
        g1[5] = (DK + 1) * SEQ;                            // dim0_stride = 129*4096 (lo32)
        g1[6] = 0;                                         // dim0_stride hi16, dim1_stride lo16
        g1[7] = 0;                                         // dim1_stride (unused, 2D)

        const i32x4 gz = {0, 0, 0, 0};
#if __clang_major__ >= 23
        const i32x8 gz8 = {0, 0, 0, 0, 0, 0, 0, 0};
        __builtin_amdgcn_tensor_load_to_lds(g0, g1, gz, gz, gz8, 0);
#else
        __builtin_amdgcn_tensor_load_to_lds(g0, g1, gz, gz, 0);
#endif
        __builtin_amdgcn_s_wait_tensorcnt(0);
    }
    __syncthreads();

    const int lane = tid & 31;   // lane -> d-quad: d = 4*lane .. 4*lane+3
    const int wv   = tid >> 5;   // 8 waves -> 8 s-positions per sweep

    const f32x4* __restrict__ xin = (const f32x4*)x;
    f32x4* __restrict__      yout = (f32x4*)out;

    for (int i = 0; i < S_CHUNK / 8; ++i) {
        const int sl = wv * (S_CHUNK / 8) + i;
        const int s  = s0 + sl;
        // 4 coefficients per thread, reused across all BH_CHUNK slices:
        // padded pitch 65 -> bank = (4*lane + j + sl) % 64, conflict-light,
        // and LDS traffic is 1/8 of global traffic anyway.
        const float c0 = coeff[(4 * lane + 0) * ROW_PITCH + sl];
        const float c1 = coeff[(4 * lane + 1) * ROW_PITCH + sl];
        const float c2 = coeff[(4 * lane + 2) * ROW_PITCH + sl];
        const float c3 = coeff[(4 * lane + 3) * ROW_PITCH + sl];
#pragma unroll
        for (int b = 0; b < BH_CHUNK; ++b) {
            // one wave = 32 lanes x float4 = one full 512B contiguous d-row
            const size_t q = ((size_t)(bh0 + b) * SEQ + s) * (DK / 4) + lane;
            f32x4 v = __builtin_nontemporal_load(&xin[q]);
            v.x *= c0;
            v.y *= c1;
            v.z *= c2;
            v.w *= c3;
            __builtin_nontemporal_store(v, &yout[q]);
        }
    }
}

extern "C" void kernel_launch(void* const* d_in, const int* in_sizes, int n_in,
                              void* d_out, int out_size, void* d_ws, size_t ws_size,
                              hipStream_t stream) {
    // setup_inputs() order: x (f32, 8*16*4096*128), token_positions (unused), R (f32)
    const float* x = (const float*)d_in[0];
    const float* R = (const float*)d_in[2];
    float* out     = (float*)d_out;

    dim3 grid(SEQ / S_CHUNK, (8 * 16) / BH_CHUNK, 1);   // 64 x 16 = 1024 blocks
    dim3 block(256, 1, 1);                              // 8 waves (wave32)
    rope_diag_scale_kernel<<<grid, block, 0, stream>>>(x, R, out);
}